// MutualAttention_37263136260605
// MI455X (gfx1250) — compile-verified
//
#include <hip/hip_runtime.h>
#include <hip/hip_bf16.h>
#include <math.h>

#define DD 512  // feature dim, fixed by the reference

typedef __attribute__((ext_vector_type(2))) float v2f;
typedef __attribute__((ext_vector_type(8))) float v8f;
typedef __attribute__((ext_vector_type(2))) int   v2i;

// --- CDNA5 async load-to-LDS path (guarded; falls back to plain loads) -------
#if defined(__AMDGCN__) && __has_builtin(__builtin_amdgcn_global_load_async_to_lds_b64) && \
    __has_builtin(__builtin_amdgcn_s_wait_asynccnt)
#define HAVE_ASYNC_LDS 1
#else
#define HAVE_ASYNC_LDS 0
#endif

__device__ __forceinline__ float fast_tanh(float x) {
#if __has_builtin(__builtin_amdgcn_tanhf)
  return __builtin_amdgcn_tanhf(x);       // v_tanh_f32 on gfx1250
#elif __has_builtin(__builtin_amdgcn_tanh_f32)
  return __builtin_amdgcn_tanh_f32(x);
#else
  return tanhf(x);
#endif
}

__device__ __forceinline__ float wave_max(float v) {
#pragma unroll
  for (int off = 16; off > 0; off >>= 1) v = fmaxf(v, __shfl_xor(v, off, 32));
  return v;
}
__device__ __forceinline__ float wave_sum(float v) {
#pragma unroll
  for (int off = 16; off > 0; off >>= 1) v += __shfl_xor(v, off, 32);
  return v;
}

// -----------------------------------------------------------------------------
// Kernel 1: pw = P[B,512] @ W[512,512], fp32, V_WMMA_F32_16X16X4_F32.
// One 16x16 output tile per wave, 8 waves (tiles) per 256-thread block.
// A frag (16x4 f32): lane&15 = M row; lane>>4 selects K pair {0,1} vs {2,3}.
// B frag (4x16 f32): lane&15 = N col; lane>>4 selects K pair (mirror of A).
// C/D (16x16 f32): VGPR v -> M = v + 8*(lane>>4), N = lane&15.
// -----------------------------------------------------------------------------
__global__ __launch_bounds__(256) void gemm_pw_wmma(const float* __restrict__ P,
                                                    const float* __restrict__ W,
                                                    float* __restrict__ PW,
                                                    int nTiles) {
  const int lane = threadIdx.x & 31;
  const int wid  = threadIdx.x >> 5;
  const int tile = blockIdx.x * 8 + wid;
  if (tile >= nTiles) return;

  const int tilesN = DD / 16;                 // 32 tiles along N
  const int tm = (tile / tilesN) * 16;
  const int tn = (tile % tilesN) * 16;

  const int r    = lane & 15;
  const int koff = (lane >> 4) << 1;          // 0 or 2

  const float* __restrict__ arow = P + (size_t)(tm + r) * DD;
  const float* __restrict__ bcol = W + tn + r;

  v8f acc = {};
  for (int k0 = 0; k0 < DD; k0 += 4) {
    v2f a = *(const v2f*)(arow + k0 + koff);              // global_load_b64
    v2f b;
    b.x = bcol[(size_t)(k0 + koff) * DD];
    b.y = bcol[(size_t)(k0 + koff + 1) * DD];
    acc = __builtin_amdgcn_wmma_f32_16x16x4_f32(
        /*neg_a=*/false, a, /*neg_b=*/false, b,
        /*c_mod=*/(short)0, acc, /*reuse_a=*/false, /*reuse_b=*/false);
  }

  const int cm = tm + (lane >> 4) * 8;
#pragma unroll
  for (int v = 0; v < 8; ++v)
    PW[(size_t)(cm + v) * DD + tn + r] = acc[v];
}

// -----------------------------------------------------------------------------
// Kernel 2: fused tanh-outer-product + dual mean + dual softmax + scale.
// One 256-thread block (8 waves) per batch element.
//   lane l owns columns {k*32+l : k=0..15} -> 16 register accumulators,
//   wave walks its 64 rows serially (x broadcast = 1 uniform LDS load/row).
// Column sums: pure register accumulation, flushed with 16 ds_add_f32/lane.
// Row sums: one 5-step shfl_xor tree per row (per 512 tanh values).
// => DS traffic ~7 ops per 512 elements; kernel is v_tanh_f32-bound.
// -----------------------------------------------------------------------------
__global__ void __launch_bounds__(256) mutual_attn_fused(const float* __restrict__ P,
                                                         const float* __restrict__ Q,
                                                         const float* __restrict__ PW,
                                                         float* __restrict__ OUTP,
                                                         float* __restrict__ OUTQ) {
  __shared__ float xs[DD], ys[DD], rowsum[DD], colsum[DD];
  __shared__ float redA[8], redB[8];

  const int b    = blockIdx.x;
  const int tid  = threadIdx.x;
  const int lane = tid & 31;
  const int wid  = tid >> 5;
  const size_t base = (size_t)b * DD;

  colsum[tid]       = 0.0f;
  colsum[tid + 256] = 0.0f;

#if HAVE_ASYNC_LDS
  {
    typedef __attribute__((address_space(1))) v2i* gp64;   // printed as "__device__"
    typedef __attribute__((address_space(3))) v2i* lp64;   // LDS ("__shared__")
    // each thread async-copies 2 floats of pw-row and q-row into LDS
    __builtin_amdgcn_global_load_async_to_lds_b64(
        (gp64)(PW + base + 2 * tid), (lp64)(&xs[2 * tid]), 0, 0);
    __builtin_amdgcn_global_load_async_to_lds_b64(
        (gp64)(Q + base + 2 * tid), (lp64)(&ys[2 * tid]), 0, 0);
    __builtin_amdgcn_s_wait_asynccnt(0);
  }
#else
  xs[tid]       = PW[base + tid];
  xs[tid + 256] = PW[base + tid + 256];
  ys[tid]       = Q[base + tid];
  ys[tid + 256] = Q[base + tid + 256];
#endif
  __syncthreads();

  // per-lane column values + column accumulators (all in registers)
  float yv[16], cs[16];
#pragma unroll
  for (int k = 0; k < 16; ++k) {
    yv[k] = ys[k * 32 + lane];
    cs[k] = 0.0f;
  }

  const int row0 = wid * 64;                    // 64 rows per wave
  for (int i = row0; i < row0 + 64; ++i) {
    const float x = xs[i];                      // uniform broadcast load
    float rs = 0.0f;
#pragma unroll
    for (int k = 0; k < 16; ++k) {
      const float v = fast_tanh(x * yv[k]);
      rs    += v;                               // row-i partial (this lane's 16 cols)
      cs[k] += v;                               // column k*32+lane partial
    }
    rs = wave_sum(rs);                          // full row sum across 512 cols
    if (lane == 0) rowsum[i] = rs;
  }
#pragma unroll
  for (int k = 0; k < 16; ++k)
    atomicAdd(&colsum[k * 32 + lane], cs[k]);   // ds_add_f32, conflict-free in-wave
  __syncthreads();

  // dual softmax: each thread owns elements {tid, tid+256} of both arrays
  const float scale = 1.0f / (float)DD;
  const float r0 = rowsum[tid] * scale, r1 = rowsum[tid + 256] * scale;
  const float c0 = colsum[tid] * scale, c1 = colsum[tid + 256] * scale;

  float mr = wave_max(fmaxf(r0, r1));
  float mc = wave_max(fmaxf(c0, c1));
  if (lane == 0) { redA[wid] = mr; redB[wid] = mc; }
  __syncthreads();
  mr = wave_max(redA[lane & 7]);                // duplicates harmless for max
  mc = wave_max(redB[lane & 7]);

  const float er0 = __expf(r0 - mr), er1 = __expf(r1 - mr);
  const float ec0 = __expf(c0 - mc), ec1 = __expf(c1 - mc);
  float sr = wave_sum(er0 + er1);
  float sc = wave_sum(ec0 + ec1);
  __syncthreads();
  if (lane == 0) { redA[wid] = sr; redB[wid] = sc; }
  __syncthreads();
  sr = wave_sum(redA[lane & 7]) * 0.25f;        // each slot counted 4x
  sc = wave_sum(redB[lane & 7]) * 0.25f;

  const float isr = 1.0f / sr, isc = 1.0f / sc;
  OUTP[base + tid]       = P[base + tid]       * (er0 * isr);
  OUTP[base + tid + 256] = P[base + tid + 256] * (er1 * isr);
  OUTQ[base + tid]       = Q[base + tid]       * (ec0 * isc);
  OUTQ[base + tid + 256] = Q[base + tid + 256] * (ec1 * isc);
}

// -----------------------------------------------------------------------------
extern "C" void kernel_launch(void* const* d_in, const int* in_sizes, int n_in,
                              void* d_out, int out_size, void* d_ws, size_t ws_size,
                              hipStream_t stream) {
  const float* p = (const float*)d_in[0];       // [B, 512]
  const float* q = (const float*)d_in[1];       // [B, 512]
  const float* w = (const float*)d_in[2];       // [512, 512]
  float* out = (float*)d_out;                   // [2, B, 512] concat

  const int B = in_sizes[0] / DD;               // 1024
  float* pw = (float*)d_ws;                     // B*512 floats (2 MB) scratch

  const int nTiles = (B / 16) * (DD / 16);      // 2048 tiles of 16x16
  const int gemmBlocks = (nTiles + 7) / 8;      // 8 waves (tiles) per block
  hipLaunchKernelGGL(gemm_pw_wmma, dim3(gemmBlocks), dim3(256), 0, stream,
                     p, w, pw, nTiles);
  hipLaunchKernelGGL(mutual_attn_fused, dim3(B), dim3(256), 0, stream,
                     p, q, pw, out, out + (size_t)B * DD);
}